// GCN_54211077210246
// MI455X (gfx1250) — compile-verified
//
#include <hip/hip_runtime.h>
#include <hip/hip_bf16.h>

typedef __attribute__((ext_vector_type(2))) float v2f;
typedef __attribute__((ext_vector_type(8))) float v8f;

#define N_NODES 50000
#define N_EDGES 800000
#define N_GRAPHS 64
#define D_IN 768
#define D_HID 256

// ---- gfx1250 async global->LDS support (guarded; falls back to sync) ------
#if defined(__has_builtin)
#if __has_builtin(__builtin_amdgcn_global_load_async_to_lds_b128) && \
    __has_builtin(__builtin_amdgcn_s_wait_asynccnt)
#define HAVE_ASYNC_LDS 1
#endif
#endif
#ifndef HAVE_ASYNC_LDS
#define HAVE_ASYNC_LDS 0
#endif

#if HAVE_ASYNC_LDS
typedef int v4i_b __attribute__((vector_size(16)));
typedef __attribute__((address_space(1))) const float* gas_cf;
typedef __attribute__((address_space(3))) float* las_f;
typedef __attribute__((address_space(1))) v4i_b* gas_v4;
typedef __attribute__((address_space(3))) v4i_b* las_v4;

__device__ __forceinline__ void async_copy16(const float* g, float* l) {
  // GLOBAL_LOAD_ASYNC_TO_LDS_B128: 16 bytes, tracked by ASYNCcnt
  __builtin_amdgcn_global_load_async_to_lds_b128(
      (gas_v4)(gas_cf)g, (las_v4)(las_f)l, 0, 0);
}
#endif

// ---------------------------------------------------------------------------
// Utility: zero a float buffer
// ---------------------------------------------------------------------------
__global__ void zero_f(float* __restrict__ p, long n) {
  long t = blockIdx.x * (long)blockDim.x + threadIdx.x;
  long stride = (long)gridDim.x * blockDim.x;
  for (; t < n; t += stride) p[t] = 0.0f;
}

// ---------------------------------------------------------------------------
// Degree accumulation over dst endpoints (deg stored in dinv buffer)
// ---------------------------------------------------------------------------
__global__ void edge_degree(const int* __restrict__ dst, float* __restrict__ deg) {
  long t = blockIdx.x * (long)blockDim.x + threadIdx.x;
  long stride = (long)gridDim.x * blockDim.x;
  for (; t < N_EDGES; t += stride) {
    __hip_atomic_fetch_add(&deg[dst[t]], 1.0f, __ATOMIC_RELAXED,
                           __HIP_MEMORY_SCOPE_AGENT);
  }
}

__global__ void finalize_dinv(float* __restrict__ dinv) {
  long t = blockIdx.x * (long)blockDim.x + threadIdx.x;
  if (t < N_NODES) dinv[t] = rsqrtf(dinv[t] + 1.0f);
}

// ---------------------------------------------------------------------------
// WMMA f32 GEMM:  C[M x 256] = A[M x K] @ B[K x 256]
//   - grid.x = M/16 row-tiles; block = 512 threads = 16 waves; each wave owns
//     one 16x16 output tile, K-loop in steps of 4 via V_WMMA_F32_16X16X4_F32
//   - A 16xK tile + double-buffered 32x256 B chunks staged in LDS with
//     GLOBAL_LOAD_ASYNC_TO_LDS_B128 (ASYNCcnt pipelined; loads complete
//     in order, so s_wait_asynccnt<=4 drains everything except the chunk
//     in flight)
//   - LDS: 16*K*4 + 2*32*256*4 bytes (112 KB at K=768) < 320 KB/WGP
// ---------------------------------------------------------------------------
template <int K>
__global__ void __launch_bounds__(512)
gemm16_wmma(const float* __restrict__ A, const float* __restrict__ B,
            float* __restrict__ C) {
  constexpr int N = 256;
  constexpr int KC = 32;
  constexpr int NCHUNK = K / KC;

  __shared__ float As[16 * K];
  __shared__ float Bs[2][KC * N];

  const int tid = threadIdx.x;  // 0..511
  const int row0 = blockIdx.x * 16;

#if HAVE_ASYNC_LDS
  // --- issue async copy of the A row-tile (float4 granularity) ---
  {
    constexpr int NV = (16 * K) / 4;  // float4 transfers
    for (int i = tid; i < NV; i += 512) {
      const int f = i * 4;
      const int r = f / K;
      const int k = f - r * K;
      async_copy16(A + (long)(row0 + r) * K + k, &As[f]);
    }
  }
  // --- issue first B chunk (rows 0..31 of B are contiguous: 32 KB) ---
  for (int i = tid; i < (KC * N) / 4; i += 512) {
    async_copy16(B + i * 4, &Bs[0][i * 4]);
  }
#else
  for (int i = tid; i < 16 * K; i += 512) {
    const int r = i / K;
    const int k = i - r * K;
    As[i] = A[(long)(row0 + r) * K + k];
  }
#endif

  const int wid = tid >> 5;
  const int lane = tid & 31;
  const int col0 = wid * 16;
  const int m = lane & 15;   // A-frag row / B-frag col
  const int hi = lane >> 4;  // 0: K pair {0,1}; 1: K pair {2,3}
  const int ksel = hi * 2;

  v8f acc = {};

  for (int ch = 0; ch < NCHUNK; ++ch) {
    const int buf = ch & 1;
#if HAVE_ASYNC_LDS
    if (ch + 1 < NCHUNK) {
      const float* src = B + (long)(ch + 1) * KC * N;
      float* dstb = &Bs[buf ^ 1][0];
      for (int i = tid; i < (KC * N) / 4; i += 512) {
        async_copy16(src + i * 4, dstb + i * 4);
      }
      __builtin_amdgcn_s_wait_asynccnt(4);  // drain all but in-flight chunk
    } else {
      __builtin_amdgcn_s_wait_asynccnt(0);
    }
    __syncthreads();
#else
    {
      const float* src = B + (long)ch * KC * N;
      for (int i = tid; i < KC * N; i += 512) Bs[buf][i] = src[i];
      __syncthreads();
    }
#endif

    const float* __restrict__ bs = &Bs[buf][0];
    for (int kk = 0; kk < KC; kk += 4) {
      const int k0 = ch * KC + kk;
      v2f a, b;
      a.x = As[m * K + k0 + ksel];
      a.y = As[m * K + k0 + ksel + 1];
      b.x = bs[(kk + ksel) * N + col0 + m];
      b.y = bs[(kk + ksel + 1) * N + col0 + m];
      acc = __builtin_amdgcn_wmma_f32_16x16x4_f32(
          /*neg_a=*/false, a, /*neg_b=*/false, b,
          /*c_mod=*/(short)0, acc, /*reuse_a=*/false, /*reuse_b=*/false);
    }
    __syncthreads();  // protect Bs[buf] before it is refilled
  }

  const int mh = hi * 8;
  for (int r = 0; r < 8; ++r) {
    C[(long)(row0 + mh + r) * N + col0 + m] = acc[r];
  }
}

// ---------------------------------------------------------------------------
// Edge aggregation: agg[dst] += dinv[src]*dinv[dst] * h[src]
// 64 threads per edge, float4 per thread, relaxed f32 atomics (L2-resident)
// ---------------------------------------------------------------------------
__global__ void edge_aggregate(const int* __restrict__ src,
                               const int* __restrict__ dst,
                               const float* __restrict__ dinv,
                               const float* __restrict__ h,
                               float* __restrict__ agg) {
  const long total = (long)N_EDGES * 64;
  long t = blockIdx.x * (long)blockDim.x + threadIdx.x;
  long stride = (long)gridDim.x * blockDim.x;
  for (; t < total; t += stride) {
    const int e = (int)(t >> 6);
    const int c = ((int)t & 63) << 2;
    const int s = src[e];
    const int d = dst[e];
    const float nrm = dinv[s] * dinv[d];
    const float4 v = *(const float4*)(h + (long)s * D_HID + c);
    float* p = agg + (long)d * D_HID + c;
    __hip_atomic_fetch_add(p + 0, nrm * v.x, __ATOMIC_RELAXED, __HIP_MEMORY_SCOPE_AGENT);
    __hip_atomic_fetch_add(p + 1, nrm * v.y, __ATOMIC_RELAXED, __HIP_MEMORY_SCOPE_AGENT);
    __hip_atomic_fetch_add(p + 2, nrm * v.z, __ATOMIC_RELAXED, __HIP_MEMORY_SCOPE_AGENT);
    __hip_atomic_fetch_add(p + 3, nrm * v.w, __ATOMIC_RELAXED, __HIP_MEMORY_SCOPE_AGENT);
  }
}

// ---------------------------------------------------------------------------
// Finalize layer: agg = [relu](agg + dinv^2 * h + bias)   (in place on agg)
// ---------------------------------------------------------------------------
__global__ void finalize_layer(float* __restrict__ agg,
                               const float* __restrict__ h,
                               const float* __restrict__ dinv,
                               const float* __restrict__ bias,
                               int do_relu) {
  const long total = (long)N_NODES * 64;
  long t = blockIdx.x * (long)blockDim.x + threadIdx.x;
  long stride = (long)gridDim.x * blockDim.x;
  for (; t < total; t += stride) {
    const int i = (int)(t >> 6);
    const int c = ((int)t & 63) << 2;
    const float di = dinv[i];
    const float d2 = di * di;
    const float4 a = *(const float4*)(agg + (long)i * D_HID + c);
    const float4 v = *(const float4*)(h + (long)i * D_HID + c);
    const float4 b = *(const float4*)(bias + c);
    float4 o;
    o.x = a.x + d2 * v.x + b.x;
    o.y = a.y + d2 * v.y + b.y;
    o.z = a.z + d2 * v.z + b.z;
    o.w = a.w + d2 * v.w + b.w;
    if (do_relu) {
      o.x = fmaxf(o.x, 0.0f); o.y = fmaxf(o.y, 0.0f);
      o.z = fmaxf(o.z, 0.0f); o.w = fmaxf(o.w, 0.0f);
    }
    *(float4*)(agg + (long)i * D_HID + c) = o;
  }
}

// ---------------------------------------------------------------------------
// Global mean pool (accumulate sums + counts via atomics)
// ---------------------------------------------------------------------------
__global__ void pool_accumulate(const float* __restrict__ h,
                                const int* __restrict__ batch,
                                float* __restrict__ pooled,
                                float* __restrict__ cnt) {
  const long total = (long)N_NODES * 64;
  long t = blockIdx.x * (long)blockDim.x + threadIdx.x;
  long stride = (long)gridDim.x * blockDim.x;
  for (; t < total; t += stride) {
    const int i = (int)(t >> 6);
    const int c = ((int)t & 63) << 2;
    const int g = batch[i];
    const float4 v = *(const float4*)(h + (long)i * D_HID + c);
    float* p = pooled + (long)g * D_HID + c;
    __hip_atomic_fetch_add(p + 0, v.x, __ATOMIC_RELAXED, __HIP_MEMORY_SCOPE_AGENT);
    __hip_atomic_fetch_add(p + 1, v.y, __ATOMIC_RELAXED, __HIP_MEMORY_SCOPE_AGENT);
    __hip_atomic_fetch_add(p + 2, v.z, __ATOMIC_RELAXED, __HIP_MEMORY_SCOPE_AGENT);
    __hip_atomic_fetch_add(p + 3, v.w, __ATOMIC_RELAXED, __HIP_MEMORY_SCOPE_AGENT);
    if (c == 0) {
      __hip_atomic_fetch_add(&cnt[g], 1.0f, __ATOMIC_RELAXED, __HIP_MEMORY_SCOPE_AGENT);
    }
  }
}

// ---------------------------------------------------------------------------
// Final linear: out[g, j] = mean_pool[g] @ Wl + bl    ([64 x 256] @ [256 x 768])
// ---------------------------------------------------------------------------
__global__ void final_linear(const float* __restrict__ pooled,
                             const float* __restrict__ cnt,
                             const float* __restrict__ Wl,
                             const float* __restrict__ bl,
                             float* __restrict__ out) {
  const int idx = blockIdx.x * blockDim.x + threadIdx.x;
  if (idx >= N_GRAPHS * D_IN) return;
  const int g = idx / D_IN;
  const int j = idx - g * D_IN;
  const float rcnt = 1.0f / fmaxf(cnt[g], 1.0f);
  float acc = 0.0f;
  const float* pr = pooled + (long)g * D_HID;
  for (int k = 0; k < D_HID; ++k) {
    acc += (pr[k] * rcnt) * Wl[(long)k * D_IN + j];
  }
  out[idx] = acc + bl[j];
}

// ---------------------------------------------------------------------------
// Host-side orchestration
// ---------------------------------------------------------------------------
extern "C" void kernel_launch(void* const* d_in, const int* in_sizes, int n_in,
                              void* d_out, int out_size, void* d_ws, size_t ws_size,
                              hipStream_t stream) {
  const float* x    = (const float*)d_in[0];
  const int*   ei   = (const int*)d_in[1];
  const int*   src  = ei;
  const int*   dst  = ei + N_EDGES;
  const int*   batch = (const int*)d_in[2];
  const float* W1 = (const float*)d_in[3];
  const float* b1 = (const float*)d_in[4];
  const float* W2 = (const float*)d_in[5];
  const float* b2 = (const float*)d_in[6];
  const float* W3 = (const float*)d_in[7];
  const float* b3 = (const float*)d_in[8];
  const float* Wl = (const float*)d_in[9];
  const float* bl = (const float*)d_in[10];
  float* out = (float*)d_out;

  // Workspace layout (floats)
  float* ws     = (float*)d_ws;
  float* dinv   = ws;                              // 50016 (deg, then dinv)
  float* pooled = dinv + 50016;                    // 64*256
  float* cnt    = pooled + (N_GRAPHS * D_HID);     // 64
  float* hA     = cnt + 64;                        // 50000*256
  float* hB     = hA + (long)N_NODES * D_HID;      // 50000*256

  const long feat_elems = (long)N_NODES * D_HID;   // 12.8M
  const int  ZB = 1024;
  const int  zgrid_feat = (int)((feat_elems + ZB - 1) / ZB);
  const long edge_work = (long)N_EDGES * 64;
  const int  egrid = (int)((edge_work + 255) / 256);
  const long node_work = (long)N_NODES * 64;
  const int  ngrid = (int)((node_work + 255) / 256);

  // --- degrees / dinv ---
  zero_f<<<(N_NODES + ZB - 1) / ZB, ZB, 0, stream>>>(dinv, N_NODES);
  edge_degree<<<(N_EDGES + 255) / 256, 256, 0, stream>>>(dst, dinv);
  finalize_dinv<<<(N_NODES + 255) / 256, 256, 0, stream>>>(dinv);

  // --- layer 1: hA = x @ W1 ; hB = relu(agg + dinv^2 hA + b1) ---
  gemm16_wmma<D_IN><<<N_NODES / 16, 512, 0, stream>>>(x, W1, hA);
  zero_f<<<zgrid_feat, ZB, 0, stream>>>(hB, feat_elems);
  edge_aggregate<<<egrid, 256, 0, stream>>>(src, dst, dinv, hA, hB);
  finalize_layer<<<ngrid, 256, 0, stream>>>(hB, hA, dinv, b1, /*relu=*/1);

  // --- layer 2: hA = hB @ W2 ; hB = relu(agg + dinv^2 hA + b2) ---
  gemm16_wmma<D_HID><<<N_NODES / 16, 512, 0, stream>>>(hB, W2, hA);
  zero_f<<<zgrid_feat, ZB, 0, stream>>>(hB, feat_elems);
  edge_aggregate<<<egrid, 256, 0, stream>>>(src, dst, dinv, hA, hB);
  finalize_layer<<<ngrid, 256, 0, stream>>>(hB, hA, dinv, b2, /*relu=*/1);

  // --- layer 3: hA = hB @ W3 ; hB = agg + dinv^2 hA + b3 (no relu) ---
  gemm16_wmma<D_HID><<<N_NODES / 16, 512, 0, stream>>>(hB, W3, hA);
  zero_f<<<zgrid_feat, ZB, 0, stream>>>(hB, feat_elems);
  edge_aggregate<<<egrid, 256, 0, stream>>>(src, dst, dinv, hA, hB);
  finalize_layer<<<ngrid, 256, 0, stream>>>(hB, hA, dinv, b3, /*relu=*/0);

  // --- global mean pool + final linear ---
  zero_f<<<(N_GRAPHS * D_HID + 64 + ZB - 1) / ZB, ZB, 0, stream>>>(
      pooled, N_GRAPHS * D_HID + 64);  // pooled and cnt are contiguous
  pool_accumulate<<<ngrid, 256, 0, stream>>>(hB, batch, pooled, cnt);
  final_linear<<<(N_GRAPHS * D_IN + 255) / 256, 256, 0, stream>>>(
      pooled, cnt, Wl, bl, out);
}